// Critic_H2G_MAAC_52175262711930
// MI455X (gfx1250) — compile-verified
//
#include <hip/hip_runtime.h>

#define N_NODES 10000
#define N_EDGES 320000
#define HID 128
#define KPAD 36   // 30 obs + 4 act, padded to multiple of 4

typedef float v2f __attribute__((ext_vector_type(2)));
typedef float v8f __attribute__((ext_vector_type(8)));

// ---------------- degree / norm ----------------
__global__ void deg_init_kernel(float* deg, int n) {
    int i = blockIdx.x * blockDim.x + threadIdx.x;
    if (i < n) deg[i] = 1.0f;   // self-loop contributes 1 to every node
}

__global__ void deg_acc_kernel(const int* __restrict__ dst, float* deg, int e) {
    int i = blockIdx.x * blockDim.x + threadIdx.x;
    if (i < e) atomicAdd(&deg[dst[i]], 1.0f);
}

__global__ void deg_rsqrt_kernel(float* deg, int n) {
    int i = blockIdx.x * blockDim.x + threadIdx.x;
    if (i < n) deg[i] = rsqrtf(deg[i]);   // deg >= 1 always (self-loops)
}

// ---------------- input packing ----------------
// xp[N][36] = [obs(30) | act(4) | 0 0]
__global__ void pack_x_kernel(const float* __restrict__ obs,
                              const float* __restrict__ act,
                              float* __restrict__ xp, int n) {
    int t = blockIdx.x * blockDim.x + threadIdx.x;
    if (t >= n * KPAD) return;
    int row = t / KPAD;
    int c   = t - row * KPAD;
    float v = 0.0f;
    if (c < 30)      v = obs[row * 30 + c];
    else if (c < 34) v = act[row * 4 + (c - 30)];
    xp[t] = v;
}

// wp[36][128] = w_g1 (34x128) zero-padded in K so the GEMM never reads OOB
__global__ void pack_w1_kernel(const float* __restrict__ w, float* __restrict__ wp) {
    int t = blockIdx.x * blockDim.x + threadIdx.x;
    if (t >= KPAD * HID) return;
    wp[t] = (t < 34 * HID) ? w[t] : 0.0f;
}

// ---------------- WMMA f32 GEMM: out[M x 128] = A[M x K] @ W[K x 128] (+bias)(+relu) ----------------
// block = 256 threads = 8 wave32; wave w owns N-columns [16w, 16w+16); grid.x tiles M by 16.
// REQUIRES: M % 16 == 0 (N_NODES = 625*16), K % 4 == 0, rows 8-byte aligned.
// Straight-line, fully unrolled, no divergence -> EXEC all-ones at every WMMA.
template <int K, bool BIAS, bool RELU>
__global__ void gemm_wmma_kernel(const float* __restrict__ A,
                                 const float* __restrict__ W,
                                 const float* __restrict__ bias,
                                 float* __restrict__ out) {
    const int lane  = threadIdx.x & 31;
    const int wave  = threadIdx.x >> 5;
    const int half  = lane >> 4;      // 0: lanes 0-15, 1: lanes 16-31
    const int l     = lane & 15;
    const int nbase = wave * 16;
    const int row0  = blockIdx.x * 16;

    v8f acc = {};

    const float* arow = A + (size_t)(row0 + l) * K;   // A row for this lane
    const float* wcol = W + nbase + l;                // this lane's B column

    #pragma unroll
    for (int k0 = 0; k0 < K; k0 += 4) {
        const int ka = k0 + half * 2;                 // ISA A layout: hi half holds K=k0+2,k0+3
        v2f a = *(const v2f*)(arow + ka);             // single global_load_b64 (8B aligned)
        v2f b;                                        // B 4x16: row ka striped over 16 lanes
        b.x = wcol[(size_t)ka * HID];
        b.y = wcol[(size_t)(ka + 1) * HID];
        acc = __builtin_amdgcn_wmma_f32_16x16x4_f32(
            /*neg_a=*/false, a, /*neg_b=*/false, b,
            /*c_mod=*/(short)0, acc, /*reuse_a=*/false, /*reuse_b=*/false);
    }

    const int c = nbase + l;
    const float bv = BIAS ? bias[c] : 0.0f;
    float* ocol = out + (size_t)(row0 + 8 * half) * HID + c;
    #pragma unroll
    for (int j = 0; j < 8; ++j) {                     // C/D layout: VGPR j -> M=j (lo) / M=8+j (hi)
        float v = acc[j] + bv;
        if (RELU) v = fmaxf(v, 0.0f);
        ocol[(size_t)j * HID] = v;
    }
}

// ---------------- GCN aggregation ----------------
// out[i] = xw[i] * dinv[i]^2   (self-loop term), float4 per lane
__global__ void selfloop_kernel(const float* __restrict__ xw,
                                const float* __restrict__ dinv,
                                float* __restrict__ out, int n) {
    int t = blockIdx.x * blockDim.x + threadIdx.x;
    if (t >= n * 32) return;
    int row = t >> 5;
    float s = dinv[row];
    s = s * s;
    float4 v = ((const float4*)xw)[t];
    float4 o;
    o.x = v.x * s; o.y = v.y * s; o.z = v.z * s; o.w = v.w * s;
    ((float4*)out)[t] = o;
}

// one wave per edge: coalesced float4 gather of the 128-float row, 4 atomic f32 adds per lane
__global__ void edge_scatter_kernel(const int* __restrict__ src, const int* __restrict__ dst,
                                    const float* __restrict__ dinv,
                                    const float* __restrict__ xw,
                                    float* __restrict__ out, int e) {
    int g    = blockIdx.x * blockDim.x + threadIdx.x;
    int eid  = g >> 5;
    int lane = g & 31;
    if (eid >= e) return;
    int s = src[eid], d = dst[eid];
    float norm = dinv[s] * dinv[d];
    float4 v = ((const float4*)(xw + (size_t)s * HID))[lane];
    float* o = out + (size_t)d * HID + lane * 4;
    atomicAdd(o + 0, v.x * norm);
    atomicAdd(o + 1, v.y * norm);
    atomicAdd(o + 2, v.z * norm);
    atomicAdd(o + 3, v.w * norm);
}

__global__ void bias_relu_kernel(float* __restrict__ x, const float* __restrict__ b, int n) {
    int t = blockIdx.x * blockDim.x + threadIdx.x;
    if (t >= n * HID) return;
    int f = t & (HID - 1);
    x[t] = fmaxf(x[t] + b[f], 0.0f);
}

// ---------------- 128 -> 1 head: one wave per node ----------------
__global__ void head_dot_kernel(const float* __restrict__ z, const float* __restrict__ w,
                                const float* __restrict__ b, float* __restrict__ q, int n) {
    int g    = blockIdx.x * blockDim.x + threadIdx.x;
    int row  = g >> 5;
    int lane = g & 31;
    if (row >= n) return;
    float4 a = ((const float4*)(z + (size_t)row * HID))[lane];
    float4 c = ((const float4*)w)[lane];
    float s = a.x * c.x + a.y * c.y + a.z * c.z + a.w * c.w;
    #pragma unroll
    for (int off = 16; off > 0; off >>= 1)
        s += __shfl_down(s, off, 32);
    if (lane == 0) q[row] = s + b[0];
}

extern "C" void kernel_launch(void* const* d_in, const int* in_sizes, int n_in,
                              void* d_out, int out_size, void* d_ws, size_t ws_size,
                              hipStream_t stream) {
    (void)in_sizes; (void)n_in; (void)out_size; (void)ws_size;

    const float* obs   = (const float*)d_in[0];
    const float* act   = (const float*)d_in[1];
    const int*   eidx  = (const int*)d_in[2];     // [2, E] flat: src then dst
    const float* w_g1  = (const float*)d_in[3];
    const float* b_g1  = (const float*)d_in[4];
    const float* w_g2  = (const float*)d_in[5];
    const float* b_g2  = (const float*)d_in[6];
    const float* w_q1a = (const float*)d_in[7];
    const float* b_q1a = (const float*)d_in[8];
    const float* w_q1b = (const float*)d_in[9];
    const float* b_q1b = (const float*)d_in[10];
    const float* w_q2a = (const float*)d_in[11];
    const float* b_q2a = (const float*)d_in[12];
    const float* w_q2b = (const float*)d_in[13];
    const float* b_q2b = (const float*)d_in[14];

    const int N = N_NODES, E = N_EDGES;
    const int* src = eidx;
    const int* dst = eidx + E;

    // workspace layout (floats)
    float* ws   = (float*)d_ws;
    float* dinv = ws;                              // 10240  (N, padded)
    float* xp   = dinv + 10240;                    // N*36
    float* wp1  = xp + (size_t)N * KPAD;           // 36*128 (padded W1), pad to 5120
    float* buf0 = wp1 + 5120;                      // N*128
    float* buf1 = buf0 + (size_t)N * HID;          // N*128
    float* buf2 = buf1 + (size_t)N * HID;          // N*128
    float* qout = (float*)d_out;                   // q1 [N] then q2 [N]

    const dim3 blk(256);
    const int gN    = (N + 255) / 256;
    const int gE    = (E + 255) / 256;
    const int gNF   = (N * HID + 255) / 256;
    const int gN32  = (N * 32 + 255) / 256;       // one wave per node
    const int gE32  = (E * 32 + 255) / 256;       // one wave per edge
    const int gGemm = N / 16;                     // 16-row M tiles (N % 16 == 0)

    // normalization
    deg_init_kernel <<<gN, blk, 0, stream>>>(dinv, N);
    deg_acc_kernel  <<<gE, blk, 0, stream>>>(dst, dinv, E);
    deg_rsqrt_kernel<<<gN, blk, 0, stream>>>(dinv, N);

    // pack inputs / padded W1
    pack_x_kernel <<<(N * KPAD + 255) / 256, blk, 0, stream>>>(obs, act, xp, N);
    pack_w1_kernel<<<(KPAD * HID + 255) / 256, blk, 0, stream>>>(w_g1, wp1);

    // GCN layer 1: xw1 = X @ W1 ; agg = norm-scatter ; h1 = relu(agg + b1)
    gemm_wmma_kernel<KPAD, false, false><<<gGemm, blk, 0, stream>>>(xp, wp1, nullptr, buf0);
    selfloop_kernel    <<<gN32, blk, 0, stream>>>(buf0, dinv, buf1, N);
    edge_scatter_kernel<<<gE32, blk, 0, stream>>>(src, dst, dinv, buf0, buf1, E);
    bias_relu_kernel   <<<gNF, blk, 0, stream>>>(buf1, b_g1, N);

    // GCN layer 2
    gemm_wmma_kernel<HID, false, false><<<gGemm, blk, 0, stream>>>(buf1, w_g2, nullptr, buf2);
    selfloop_kernel    <<<gN32, blk, 0, stream>>>(buf2, dinv, buf0, N);
    edge_scatter_kernel<<<gE32, blk, 0, stream>>>(src, dst, dinv, buf2, buf0, E);
    bias_relu_kernel   <<<gNF, blk, 0, stream>>>(buf0, b_g2, N);

    // Q heads: z = relu(h2 @ Wa + ba) ; q = z @ wb + bb
    gemm_wmma_kernel<HID, true, true><<<gGemm, blk, 0, stream>>>(buf0, w_q1a, b_q1a, buf1);
    head_dot_kernel <<<gN32, blk, 0, stream>>>(buf1, w_q1b, b_q1b, qout, N);
    gemm_wmma_kernel<HID, true, true><<<gGemm, blk, 0, stream>>>(buf0, w_q2a, b_q2a, buf1);
    head_dot_kernel <<<gN32, blk, 0, stream>>>(buf1, w_q2b, b_q2b, qout + N, N);
}